// GINConv_28716151341439
// MI455X (gfx1250) — compile-verified
//
#include <hip/hip_runtime.h>

typedef __attribute__((ext_vector_type(2))) float v2f;
typedef __attribute__((ext_vector_type(8))) float v8f;

#define NN     50000
#define DD     96
#define NE     800000
#define KTILES 24          // 96 / 4
#define NTILES 6           // 96 / 16
#define MTILES 3125        // 50000 / 16  (16-row WMMA tiles)
#define STILES 1563        // ceil(MTILES / 2) -- 32-row supertiles per wave
#define WAVES  4
#define PSTR   224         // LDS stride between K-row pairs (224 mod 64 = 32 -> conflict-free)
#define HSTR   100         // LDS stride for h staging (transpose-read conflict-free)

// ---- kernel 1: rst = feat (eps = 0) ----
__global__ __launch_bounds__(256) void k_copy(const float* __restrict__ src,
                                              float* __restrict__ dst, int n4) {
    int i = blockIdx.x * blockDim.x + threadIdx.x;
    if (i < n4) ((float4*)dst)[i] = ((const float4*)src)[i];
}

// ---- kernel 2: segment-sum of source features into rst via f32 atomics ----
__global__ __launch_bounds__(256) void k_scatter(const float* __restrict__ feat,
                                                 const int*   __restrict__ esrc,
                                                 const int*   __restrict__ edst,
                                                 float*       __restrict__ rst) {
    int w    = (blockIdx.x * blockDim.x + threadIdx.x) >> 5;   // one wave per edge
    int lane = threadIdx.x & 31;
    if (w >= NE) return;
    int s = esrc[w];
    int d = edst[w];
    const float* fs = feat + (size_t)s * DD;
    float*       rd = rst  + (size_t)d * DD;
#pragma unroll
    for (int j = 0; j < 3; ++j) {
        float v = fs[lane + 32 * j];
        __hip_atomic_fetch_add(&rd[lane + 32 * j], v,
                               __ATOMIC_RELAXED, __HIP_MEMORY_SCOPE_AGENT);
    }
}

// ---- kernel 3: out = relu(rst @ W1 + b1) @ W2 + b2, fused, f32 WMMA ----
// Each wave owns a 32-row supertile (2 x 16-row WMMA tiles) so every B-fragment
// ds_load_b64 feeds two v_wmma_f32_16x16x4_f32.
__global__ __launch_bounds__(32 * WAVES) void k_mlp(const float* __restrict__ rst,
                                                    const float* __restrict__ W1,
                                                    const float* __restrict__ b1,
                                                    const float* __restrict__ W2,
                                                    const float* __restrict__ b2,
                                                    float* __restrict__ out) {
    __shared__ float sW1p[48 * PSTR];        // K-row-pair interleaved W1
    __shared__ float sW2p[48 * PSTR];        // K-row-pair interleaved W2
    __shared__ float sB1[DD];
    __shared__ float sB2[DD];
    __shared__ float sH[WAVES * 2][16 * HSTR];   // per-(wave,tile) h staging

    int tid = threadIdx.x;
    // stage weights: sWp[p*PSTR + col*2 + r] = W[(2p + r)*96 + col]
    for (int i = tid; i < DD * DD / 4; i += 32 * WAVES) {
        int e   = i * 4;
        int row = e / DD, col = e - row * DD;          // 4 elems share one row
        int p = row >> 1, rr = row & 1;
        float4 w1 = ((const float4*)W1)[i];
        float4 w2 = ((const float4*)W2)[i];
        float* d1 = &sW1p[p * PSTR + col * 2 + rr];
        float* d2 = &sW2p[p * PSTR + col * 2 + rr];
        d1[0] = w1.x; d1[2] = w1.y; d1[4] = w1.z; d1[6] = w1.w;
        d2[0] = w2.x; d2[2] = w2.y; d2[4] = w2.z; d2[6] = w2.w;
    }
    if (tid < DD) { sB1[tid] = b1[tid]; sB2[tid] = b2[tid]; }
    __syncthreads();

    int wv   = tid >> 5;
    int lane = tid & 31;
    int half = lane >> 4;    // 0: lanes 0-15, 1: lanes 16-31
    int l16  = lane & 15;

    int st = blockIdx.x * WAVES + wv;
    if (st >= STILES) return;
    int  tile0 = 2 * st;
    bool has2  = (tile0 + 1) < MTILES;
    int  tile1 = has2 ? (tile0 + 1) : tile0;   // clamp: duplicate compute, stores guarded

    // A fragments: 16x4 f32 layout -> lane holds A[l16][half*2 + {0,1}]
    const float* arow0 = rst + (size_t)(tile0 * 16 + l16) * DD + half * 2;
    const float* arow1 = rst + (size_t)(tile1 * 16 + l16) * DD + half * 2;
    v2f a0[KTILES], a1[KTILES];
#pragma unroll
    for (int k = 0; k < KTILES; ++k) {
        a0[k] = *(const v2f*)(arow0 + k * 4);
        a1[k] = *(const v2f*)(arow1 + k * 4);
    }

    float* h0 = &sH[wv * 2 + 0][0];
    float* h1 = &sH[wv * 2 + 1][0];

    // ---------------- GEMM1: h = relu(A*W1 + b1) ----------------
    for (int n = 0; n < NTILES; ++n) {
        v8f acc0 = {}, acc1 = {};
        int c2 = (n * 16 + l16) * 2;
#pragma unroll
        for (int k = 0; k < KTILES; ++k) {
            // one b64: B[kb][col], B[kb+1][col] with kb = k*4 + half*2
            v2f b = *(const v2f*)&sW1p[(k * 2 + half) * PSTR + c2];
            acc0 = __builtin_amdgcn_wmma_f32_16x16x4_f32(
                       false, a0[k], false, b, (short)0, acc0, false, false);
            acc1 = __builtin_amdgcn_wmma_f32_16x16x4_f32(
                       false, a1[k], false, b, (short)0, acc1, false, false);
        }
        float bias = sB1[n * 16 + l16];
#pragma unroll
        for (int r = 0; r < 8; ++r) {      // C layout: VGPR r -> row r + half*8
            float v0 = acc0[r] + bias; v0 = v0 > 0.0f ? v0 : 0.0f;
            float v1 = acc1[r] + bias; v1 = v1 > 0.0f ? v1 : 0.0f;
            h0[(half * 8 + r) * HSTR + n * 16 + l16] = v0;
            h1[(half * 8 + r) * HSTR + n * 16 + l16] = v1;
        }
    }

    // cross-lane LDS transpose hazard: wait for this wave's DS stores
    asm volatile("s_wait_dscnt 0x0" ::: "memory");

    // reload A fragments from the h tiles (row-major in LDS)
#pragma unroll
    for (int k = 0; k < KTILES; ++k) {
        a0[k] = *(const v2f*)(h0 + l16 * HSTR + k * 4 + half * 2);
        a1[k] = *(const v2f*)(h1 + l16 * HSTR + k * 4 + half * 2);
    }

    // ---------------- GEMM2: out = h*W2 + b2 ----------------
    float* o0 = out + (size_t)tile0 * 16 * DD;
    float* o1 = out + (size_t)tile1 * 16 * DD;
    for (int n = 0; n < NTILES; ++n) {
        v8f acc0 = {}, acc1 = {};
        int c2 = (n * 16 + l16) * 2;
#pragma unroll
        for (int k = 0; k < KTILES; ++k) {
            v2f b = *(const v2f*)&sW2p[(k * 2 + half) * PSTR + c2];
            acc0 = __builtin_amdgcn_wmma_f32_16x16x4_f32(
                       false, a0[k], false, b, (short)0, acc0, false, false);
            acc1 = __builtin_amdgcn_wmma_f32_16x16x4_f32(
                       false, a1[k], false, b, (short)0, acc1, false, false);
        }
        float bias = sB2[n * 16 + l16];
#pragma unroll
        for (int r = 0; r < 8; ++r)
            o0[(half * 8 + r) * DD + n * 16 + l16] = acc0[r] + bias;
        if (has2) {
#pragma unroll
            for (int r = 0; r < 8; ++r)
                o1[(half * 8 + r) * DD + n * 16 + l16] = acc1[r] + bias;
        }
    }
}

extern "C" void kernel_launch(void* const* d_in, const int* in_sizes, int n_in,
                              void* d_out, int out_size, void* d_ws, size_t ws_size,
                              hipStream_t stream) {
    const float* feat = (const float*)d_in[0];
    const float* W1   = (const float*)d_in[1];
    const float* b1   = (const float*)d_in[2];
    const float* W2   = (const float*)d_in[3];
    const float* b2   = (const float*)d_in[4];
    const int*   esrc = (const int*)d_in[5];
    const int*   edst = (const int*)d_in[6];
    float* out = (float*)d_out;
    float* rst = (float*)d_ws;          // [NN * DD] f32 accumulator

    int n4 = NN * DD / 4;
    k_copy<<<(n4 + 255) / 256, 256, 0, stream>>>(feat, rst, n4);

    long long sc_threads = (long long)NE * 32;
    k_scatter<<<(int)((sc_threads + 255) / 256), 256, 0, stream>>>(feat, esrc, edst, rst);

    k_mlp<<<(STILES + WAVES - 1) / WAVES, 32 * WAVES, 0, stream>>>(rst, W1, b1, W2, b2, out);
}